// PatchletsExtractor_6957847020166
// MI455X (gfx1250) — compile-verified
//
#include <hip/hip_runtime.h>

typedef __attribute__((ext_vector_type(2))) float v2f;
typedef __attribute__((ext_vector_type(8))) float v8f;

#define NP 2048        // points per frame
#define KK 16          // neighbors
#define TCLIP 64       // frames per clip (prev-frame duplication at t==0)
#define PASS_COLS 512  // database columns staged in LDS per pass

__device__ __forceinline__ bool lex_less(float d1, int i1, float d2, int i2) {
  return (d1 < d2) || (d1 == d2 && i1 < i2);
}

// ---------------------------------------------------------------------------
// Kernel 1: per frame f, per 16-query tile: scores via V_WMMA_F32_16X16X4_F32,
// streaming register top-16 per query with (dist, idx) lexicographic order.
// d2 = |q|^2 + |b|^2 - 2 q.b ; WMMA computes (q,1)·(-2b,|b|^2) = d2 - |q|^2,
// which preserves per-row ordering; |q|^2 is added to the 16 winners only.
// ---------------------------------------------------------------------------
__global__ void __launch_bounds__(256)
knn_topk_kernel(const float* __restrict__ pts,   // [F, NP, 3]
                float* __restrict__ out_idx,     // [F, NP, KK] (float-encoded)
                float* __restrict__ out_dist)    // [F, NP, KK]
{
  __shared__ float smem[16 * PASS_COLS];  // 32 KB; reused for candidate merge

  const int mtile = blockIdx.x;               // 0..127
  const int f     = blockIdx.y;               // 0..F-1
  const int fprev = ((f % TCLIP) == 0) ? f : (f - 1);
  const float* x1 = pts + (size_t)f * NP * 3;      // queries (current frame)
  const float* x2 = pts + (size_t)fprev * NP * 3;  // database (previous frame)
  const int q0 = mtile * 16;

  const int tid  = threadIdx.x;
  const int wave = tid >> 5;      // 8 waves (wave32)
  const int lane = tid & 31;
  const int lm   = lane & 15;
  const bool hi  = lane >= 16;

  // A tile (16x4 f32, ISA layout): lanes 0-15 VGPR{0,1} = K{0,1} = (qx,qy);
  // lanes 16-31 VGPR{0,1} = K{2,3} = (qz, 1.0)
  const float* qp = x1 + (size_t)(q0 + lm) * 3;
  const float qx = qp[0], qy = qp[1], qz = qp[2];
  v2f A;
  A.x = hi ? qz : qx;
  A.y = hi ? 1.0f : qy;

  const int qi = tid >> 4;  // phase-2 query row 0..15
  const int ch = tid & 15;  // phase-2 column chunk 0..15 (32 cols/chunk/pass)

  float bd[KK]; int bi[KK];
#pragma unroll
  for (int r = 0; r < KK; ++r) { bd[r] = 3.402823466e38f; bi[r] = 0x7fffffff; }

  for (int pass = 0; pass < NP / PASS_COLS; ++pass) {
    // ---- phase 1: 8 waves x 4 tiles -> smem scores[16][512]
#pragma unroll
    for (int i = 0; i < 4; ++i) {
      const int c0 = pass * PASS_COLS + (wave * 4 + i) * 16;
      const float* bp = x2 + (size_t)(c0 + lm) * 3;
      const float bx = bp[0], by = bp[1], bz = bp[2];
      const float bn2 = bx * bx + by * by + bz * bz;
      // B tile (4x16 f32): column = db point, packed (-2bx,-2by,-2bz,|b|^2)
      v2f B;
      B.x = hi ? (-2.0f * bz) : (-2.0f * bx);
      B.y = hi ? bn2 : (-2.0f * by);
      v8f C = {};
      C = __builtin_amdgcn_wmma_f32_16x16x4_f32(false, A, false, B,
                                                (short)0, C, false, false);
      // D layout: VGPR r -> (M = r + (hi?8:0), N = lm)
      const int colL = (wave * 4 + i) * 16 + lm;
      const int mb = hi ? 8 : 0;
#pragma unroll
      for (int r = 0; r < 8; ++r)
        smem[(mb + r) * PASS_COLS + colL] = C[r];
    }
    __syncthreads();

    // ---- phase 2: streaming register top-16 per (query, chunk)
    const float* row = smem + qi * PASS_COLS + ch * 32;
    const int gbase = pass * PASS_COLS + ch * 32;
    for (int j = 0; j < 32; ++j) {
      const float v = row[j];
      const int id = gbase + j;
      if (lex_less(v, id, bd[KK - 1], bi[KK - 1])) {
        bd[KK - 1] = v; bi[KK - 1] = id;
        bool done = false;
#pragma unroll
        for (int r = KK - 1; r >= 1; --r) {
          if (!done) {
            if (lex_less(bd[r], bi[r], bd[r - 1], bi[r - 1])) {
              const float td = bd[r]; bd[r] = bd[r - 1]; bd[r - 1] = td;
              const int ti = bi[r]; bi[r] = bi[r - 1]; bi[r - 1] = ti;
            } else done = true;
          }
        }
      }
    }
    __syncthreads();
  }

  // ---- merge: reuse smem: 16 queries x 16 chunks x 16 candidates
  float* candD = smem;                               // 4096 floats
  int* candI = (int*)(smem + 16 * 16 * KK);          // 4096 ints
#pragma unroll
  for (int r = 0; r < KK; ++r) {
    candD[(qi * 16 + ch) * KK + r] = bd[r];
    candI[(qi * 16 + ch) * KK + r] = bi[r];
  }
  __syncthreads();

  if (tid < 16) {
    const int qq = tid;
    float md[KK]; int mi[KK];
#pragma unroll
    for (int r = 0; r < KK; ++r) { md[r] = 3.402823466e38f; mi[r] = 0x7fffffff; }
    const float* cd = candD + qq * 16 * KK;
    const int* ci = candI + qq * 16 * KK;
    for (int j = 0; j < 16 * KK; ++j) {
      const float v = cd[j]; const int id = ci[j];
      if (lex_less(v, id, md[KK - 1], mi[KK - 1])) {
        md[KK - 1] = v; mi[KK - 1] = id;
        bool done = false;
#pragma unroll
        for (int r = KK - 1; r >= 1; --r) {
          if (!done) {
            if (lex_less(md[r], mi[r], md[r - 1], mi[r - 1])) {
              const float td = md[r]; md[r] = md[r - 1]; md[r - 1] = td;
              const int ti = mi[r]; mi[r] = mi[r - 1]; mi[r - 1] = ti;
            } else done = true;
          }
        }
      }
    }
    const float* qv = x1 + (size_t)(q0 + qq) * 3;
    const float qn2 = qv[0] * qv[0] + qv[1] * qv[1] + qv[2] * qv[2];
    const size_t ob = ((size_t)f * NP + (size_t)(q0 + qq)) * KK;
#pragma unroll
    for (int r = 0; r < KK; ++r) {
      out_idx[ob + r]  = (float)mi[r];
      out_dist[ob + r] = md[r] + qn2;
    }
  }
}

// ---------------------------------------------------------------------------
// Kernel 2: sequential patchlet chain; per-position q independent:
//   s_0 = q ; patchlets[f][q][:] = idx[f][s_f][:] ; s_{f+1} = idx[f][s_f][0]
// ---------------------------------------------------------------------------
__global__ void __launch_bounds__(256)
patchlet_chain_kernel(const float* __restrict__ idxs,  // [F, NP, KK] float-enc
                      float* __restrict__ patch,       // [F, NP, KK]
                      int F)
{
  const int q = blockIdx.x * blockDim.x + threadIdx.x;
  if (q >= NP) return;
  int s = q;
  for (int f = 0; f < F; ++f) {
    const float* row = idxs + ((size_t)f * NP + (size_t)s) * KK;
    const float r0 = row[0];
#pragma unroll
    for (int r = 0; r < KK; ++r)
      patch[((size_t)f * NP + (size_t)q) * KK + r] = row[r];
    s = (int)r0;
  }
}

// ---------------------------------------------------------------------------
// Kernel 3: gather patchlet_points [F,NP,KK,3] and patchlet_feats [F,NP,KK,16]
// ---------------------------------------------------------------------------
__global__ void __launch_bounds__(256)
gather_kernel(const float* __restrict__ pts,    // [F, NP, 3]
              const float* __restrict__ feats,  // [F, NP, 16]
              const float* __restrict__ patch,  // [F, NP, KK] float-encoded
              float* __restrict__ out_pts,      // [F, NP, KK, 3]
              float* __restrict__ out_feats,    // [F, NP, KK, 16]
              long total)                       // F*NP*KK
{
  const long g = (long)blockIdx.x * blockDim.x + threadIdx.x;
  if (g >= total) return;
  const long fq = g >> 4;          // f*NP + q
  const long f = fq / NP;
  const int p = (int)patch[g];

  const float* sp = pts + ((size_t)f * NP + (size_t)p) * 3;
  float* op = out_pts + (size_t)g * 3;
  op[0] = sp[0]; op[1] = sp[1]; op[2] = sp[2];

  const float4* sf = (const float4*)(feats + ((size_t)f * NP + (size_t)p) * 16);
  float4* of = (float4*)(out_feats + (size_t)g * 16);
  of[0] = sf[0]; of[1] = sf[1]; of[2] = sf[2]; of[3] = sf[3];
}

// ---------------------------------------------------------------------------
extern "C" void kernel_launch(void* const* d_in, const int* in_sizes, int n_in,
                              void* d_out, int out_size, void* d_ws,
                              size_t ws_size, hipStream_t stream) {
  const float* pts   = (const float*)d_in[0];   // [4,64,2048,3]
  const float* feats = (const float*)d_in[1];   // [4,64,2048,16]
  const int F = in_sizes[0] / (NP * 3);         // 256 flattened frames

  float* out = (float*)d_out;
  const size_t FNK = (size_t)F * NP * KK;
  float* out_idx   = out;                       // idxs            [F,NP,KK]
  float* out_dist  = out + FNK;                 // distances       [F,NP,KK]
  float* out_patch = out + 2 * FNK;             // patchlets       [F,NP,KK]
  float* out_pts   = out + 3 * FNK;             // patchlet_points [F,NP,KK,3]
  float* out_feat  = out + 3 * FNK + FNK * 3;   // patchlet_feats  [F,NP,KK,16]

  // 1) per-frame KNN + top-16 (WMMA scoring)
  dim3 g1(NP / 16, F);
  knn_topk_kernel<<<g1, 256, 0, stream>>>(pts, out_idx, out_dist);

  // 2) sequential chain across frames (parallel over positions)
  patchlet_chain_kernel<<<NP / 256, 256, 0, stream>>>(out_idx, out_patch, F);

  // 3) gathers
  const long total = (long)F * NP * KK;
  gather_kernel<<<(unsigned)((total + 255) / 256), 256, 0, stream>>>(
      pts, feats, out_patch, out_pts, out_feat, total);
}